// MemNet_46024869543936
// MI455X (gfx1250) — compile-verified
//
#include <hip/hip_runtime.h>
#include <hip/hip_bf16.h>
#include <math.h>

// Sizes from the reference
#define Bsz 256
#define Lsz 512
#define Tsz 5
#define Dsz 300
#define Csz 3
#define NHOPS 3

#define CHUNK 32                 // ext rows staged to LDS per iteration
#define NCHUNK (Lsz / CHUNK)     // 16
#define XFERS (CHUNK * Dsz / 4)  // 16B transfers per chunk = 2400

typedef __attribute__((ext_vector_type(2))) float v2f;
typedef __attribute__((ext_vector_type(8))) float v8f;

// ---------------------------------------------------------------------------
// K0: v_aspect seed vector: vec[b,d] = mean over valid t of emb[target_x[b,t], d]
// ---------------------------------------------------------------------------
__global__ void v_aspect_kernel(const int* __restrict__ target_x,
                                const int* __restrict__ target_len,
                                const float* __restrict__ emb,
                                float* __restrict__ vec) {
  int b = blockIdx.x;
  int tl = target_len[b];  // 1..T
  for (int d = threadIdx.x; d < Dsz; d += blockDim.x) {
    float acc = 0.f;
    #pragma unroll
    for (int t = 0; t < Tsz; ++t) {
      if (t < tl) acc += emb[(size_t)target_x[b * Tsz + t] * Dsz + d];
    }
    vec[b * Dsz + d] = acc / (float)tl;
  }
}

// ---------------------------------------------------------------------------
// K1: ext_memory[b,l,:] = emb[context_x[b,l],:] * v_loc(b,l)
//     mem_score[b,l]    = ext_memory[b,l,:] . w_mem   (hop-invariant)
// ---------------------------------------------------------------------------
__global__ void build_ext_kernel(const int* __restrict__ context_x,
                                 const int* __restrict__ context_len,
                                 const int* __restrict__ target_loc,
                                 const float* __restrict__ emb,
                                 const float* __restrict__ w_mem,  // attn_w[0:D]
                                 float* __restrict__ ext,
                                 float* __restrict__ mem_score) {
  int l = blockIdx.x;
  int b = blockIdx.y;
  int tid = threadIdx.x;  // 128 threads
  __shared__ float sred[128];

  int tok = context_x[b * Lsz + l];
  float dist = fabsf((float)(l - target_loc[b]));
  float vloc = 1.0f - dist / (float)context_len[b];

  const float* erow = emb + (size_t)tok * Dsz;
  float* xrow = ext + ((size_t)b * Lsz + l) * Dsz;

  float p = 0.f;
  for (int d = tid; d < Dsz; d += 128) {
    float v = erow[d] * vloc;
    xrow[d] = v;
    p += v * w_mem[d];
  }
  sred[tid] = p;
  __syncthreads();
  for (int s = 64; s > 0; s >>= 1) {
    if (tid < s) sred[tid] += sred[tid + s];
    __syncthreads();
  }
  if (tid == 0) mem_score[b * Lsz + l] = sred[0];
}

// ---------------------------------------------------------------------------
// K2: linear_out = vec @ lin_w + lin_b  via V_WMMA_F32_16X16X4_F32
//     One wave per 16x16 tile; K loop = 75 steps of 4. Branchless N-edge:
//     clamped column + 0/1 mask so the hot loop has no exec manipulation.
// ---------------------------------------------------------------------------
__global__ void gemm_lin_kernel(const float* __restrict__ vec,    // B x D
                                const float* __restrict__ lin_w,  // D x D row-major [k][n]
                                const float* __restrict__ lin_b,  // D
                                float* __restrict__ lin_out) {    // B x D
  int tile_m = blockIdx.x;  // 0..15
  int tile_n = blockIdx.y;  // 0..18
  int lane = threadIdx.x;   // 0..31
  int half = lane >> 4;
  int l16 = lane & 15;

  int row = tile_m * 16 + l16;   // M index for this lane's A fragment
  int ncol = tile_n * 16 + l16;  // N index for this lane's B fragment
  bool nvalid = ncol < Dsz;
  int bcol = nvalid ? ncol : (Dsz - 1);       // clamped, always in-bounds
  float bmask = nvalid ? 1.0f : 0.0f;         // zero out the pad lanes

  const float* arow = vec + row * Dsz;
  v8f acc = {};
  for (int k0 = 0; k0 < Dsz; k0 += 4) {
    int ka = k0 + half * 2;
    v2f a, bfrag;
    a.x = arow[ka + 0];
    a.y = arow[ka + 1];
    bfrag.x = lin_w[(ka + 0) * Dsz + bcol] * bmask;
    bfrag.y = lin_w[(ka + 1) * Dsz + bcol] * bmask;
    acc = __builtin_amdgcn_wmma_f32_16x16x4_f32(
        /*neg_a=*/false, a, /*neg_b=*/false, bfrag,
        /*c_mod=*/(short)0, acc, /*reuse_a=*/false, /*reuse_b=*/false);
  }

  if (nvalid) {
    float bias = lin_b[ncol];
    #pragma unroll
    for (int j = 0; j < 8; ++j) {
      int m = tile_m * 16 + j + half * 8;
      lin_out[m * Dsz + ncol] = acc[j] + bias;
    }
  }
}

// ---------------------------------------------------------------------------
// K3: per-b fused attention hop. ext[b] rows are staged through LDS in 32-row
// chunks with async global->LDS copies (gfx1250 GLOBAL_LOAD_ASYNC_TO_LDS_B128,
// tracked by ASYNCcnt), prefetching the next chunk while computing.
// ---------------------------------------------------------------------------
__global__ void attn_kernel(const float* __restrict__ ext,        // B x L x D
                            const float* __restrict__ mem_score,  // B x L
                            const float* __restrict__ lin_out,    // B x D
                            const float* __restrict__ w_vec,      // attn_w[D:2D]
                            const float* __restrict__ attn_b,     // 1
                            const int* __restrict__ context_len,  // B
                            float* __restrict__ vec) {            // B x D (updated)
  int b = blockIdx.x;
  int tid = threadIdx.x;  // 256 threads
  __shared__ float sred[256];
  __shared__ float salpha[Lsz];
  __shared__ __attribute__((aligned(16))) float stile[CHUNK * Dsz];

  // vdot = vec[b,:] . w_vec
  float p = 0.f;
  for (int d = tid; d < Dsz; d += 256) p += vec[b * Dsz + d] * w_vec[d];
  sred[tid] = p;
  __syncthreads();
  for (int s = 128; s > 0; s >>= 1) {
    if (tid < s) sred[tid] += sred[tid + s];
    __syncthreads();
  }
  float vdot = sred[0] + attn_b[0];
  __syncthreads();

  int clen = context_len[b];

  // scores + running max (each thread covers 2 of the 512 positions)
  float mymax = -3.0e38f;
  #pragma unroll
  for (int i = 0; i < 2; ++i) {
    int l = tid + i * 256;
    float s = (l < clen) ? tanhf(mem_score[b * Lsz + l] + vdot) : -1.0e9f;
    salpha[l] = s;
    mymax = fmaxf(mymax, s);
  }
  sred[tid] = mymax;
  __syncthreads();
  for (int s = 128; s > 0; s >>= 1) {
    if (tid < s) sred[tid] = fmaxf(sred[tid], sred[tid + s]);
    __syncthreads();
  }
  float mx = sred[0];
  __syncthreads();

  // exp + sum
  float psum = 0.f;
  #pragma unroll
  for (int i = 0; i < 2; ++i) {
    int l = tid + i * 256;
    float e = __expf(salpha[l] - mx);
    salpha[l] = e;
    psum += e;
  }
  sred[tid] = psum;
  __syncthreads();
  for (int s = 128; s > 0; s >>= 1) {
    if (tid < s) sred[tid] += sred[tid + s];
    __syncthreads();
  }
  float inv = 1.0f / sred[0];
  __syncthreads();
  #pragma unroll
  for (int i = 0; i < 2; ++i) salpha[tid + i * 256] *= inv;
  __syncthreads();

  // attn_out[d] = sum_l alpha[l] * ext[b,l,d], chunked through LDS.
  // Generic->u32 truncation of a __shared__ pointer yields the allocation-
  // relative LDS byte offset (low 32 bits of the LDS aperture address),
  // which is what the async-load VDST operand expects (HW adds LDS_BASE).
  unsigned lds_base = (unsigned)(unsigned long long)(uintptr_t)stile;
  const float* gbase = ext + (size_t)b * Lsz * Dsz;

  float acc0 = 0.f, acc1 = 0.f;  // this thread's d and d+256 accumulators
  for (int chunk = 0; chunk < NCHUNK; ++chunk) {
    const float* gchunk = gbase + (size_t)chunk * CHUNK * Dsz;  // contiguous 38400B
    // Issue async 16B copies: per-lane LDS dst offset + 32-bit voffset, uniform SGPR base.
    for (int i = tid; i < XFERS; i += 256) {
      unsigned la = lds_base + (unsigned)(i * 16);
      unsigned vo = (unsigned)(i * 16);
      asm volatile("global_load_async_to_lds_b128 %0, %1, %2"
                   :: "v"(la), "v"(vo), "s"(gchunk)
                   : "memory");
    }
    // Prefetch next chunk into L2/L0 while this one lands & is consumed.
    if (chunk + 1 < NCHUNK) {
      const char* nxt = (const char*)(gchunk + (size_t)CHUNK * Dsz);
      __builtin_prefetch(nxt + tid * 128, 0, 0);
    }
    asm volatile("s_wait_asynccnt 0x0" ::: "memory");
    __syncthreads();

    const float* al = &salpha[chunk * CHUNK];
    #pragma unroll 4
    for (int lr = 0; lr < CHUNK; ++lr) {
      float a = al[lr];
      acc0 += a * stile[lr * Dsz + tid];
      if (tid + 256 < Dsz) acc1 += a * stile[lr * Dsz + tid + 256];
    }
    __syncthreads();  // before overwriting stile with the next chunk
  }

  vec[b * Dsz + tid] = acc0 + lin_out[b * Dsz + tid];
  if (tid + 256 < Dsz)
    vec[b * Dsz + tid + 256] = acc1 + lin_out[b * Dsz + tid + 256];
}

// ---------------------------------------------------------------------------
// K4: logits = vec @ out_w + out_b   (256 x 300 @ 300 x 3 -> tiny)
// ---------------------------------------------------------------------------
__global__ void out_kernel(const float* __restrict__ vec,
                           const float* __restrict__ out_w,  // D x C
                           const float* __restrict__ out_b,  // C
                           float* __restrict__ logits) {     // B x C
  int b = blockIdx.x * blockDim.x + threadIdx.x;
  if (b >= Bsz) return;
  #pragma unroll
  for (int c = 0; c < Csz; ++c) {
    float acc = out_b[c];
    for (int d = 0; d < Dsz; ++d) acc += vec[b * Dsz + d] * out_w[d * Csz + c];
    logits[b * Csz + c] = acc;
  }
}

extern "C" void kernel_launch(void* const* d_in, const int* in_sizes, int n_in,
                              void* d_out, int out_size, void* d_ws, size_t ws_size,
                              hipStream_t stream) {
  const int*   context_x   = (const int*)d_in[0];
  const int*   context_len = (const int*)d_in[1];
  const int*   target_x    = (const int*)d_in[2];
  const int*   target_len  = (const int*)d_in[3];
  const int*   target_loc  = (const int*)d_in[4];
  const float* emb         = (const float*)d_in[5];
  const float* attn_w      = (const float*)d_in[6];  // (2D,1): w_mem then w_vec
  const float* attn_b      = (const float*)d_in[7];
  const float* lin_w       = (const float*)d_in[8];
  const float* lin_b       = (const float*)d_in[9];
  const float* out_w       = (const float*)d_in[10];
  const float* out_b       = (const float*)d_in[11];
  float* logits = (float*)d_out;

  const float* w_mem = attn_w;
  const float* w_vec = attn_w + Dsz;

  // Workspace carve-up
  char* ws = (char*)d_ws;
  float* ext       = (float*)ws;                                      // B*L*D
  float* mem_score = (float*)(ws + (size_t)Bsz * Lsz * Dsz * 4);      // B*L
  float* vec       = mem_score + (size_t)Bsz * Lsz;                   // B*D
  float* lin_out   = vec + (size_t)Bsz * Dsz;                         // B*D

  // K0: seed vector
  v_aspect_kernel<<<Bsz, 128, 0, stream>>>(target_x, target_len, emb, vec);

  // K1: hop-invariant ext_memory + mem_score (one pass over the gather)
  build_ext_kernel<<<dim3(Lsz, Bsz), 128, 0, stream>>>(
      context_x, context_len, target_loc, emb, w_mem, ext, mem_score);

  // Hops: WMMA GEMM for lin layer, fused softmax+attention update
  dim3 gemm_grid(Bsz / 16, (Dsz + 15) / 16);  // 16 x 19 tiles
  for (int hop = 0; hop < NHOPS; ++hop) {
    gemm_lin_kernel<<<gemm_grid, 32, 0, stream>>>(vec, lin_w, lin_b, lin_out);
    attn_kernel<<<Bsz, 256, 0, stream>>>(ext, mem_score, lin_out, w_vec, attn_b,
                                         context_len, vec);
  }

  // K4: output projection
  out_kernel<<<1, Bsz, 0, stream>>>(vec, out_w, out_b, logits);
}